// RandomSpecAugment_18854906429551
// MI455X (gfx1250) — compile-verified
//
#include <hip/hip_runtime.h>

// SpecAugment elementwise kernel for MI455X (gfx1250).
// out[b,0,f,t] = x * band * (keep[b]/(1-p)) + noise * 0.05
// Memory-bound: 12 B/elem, ~402 MB total -> ~17 us floor @ 23.3 TB/s.
// Strategy: one block per (b,f) row; uniform scalar params per block;
// float4 (b128) non-temporal loads/stores to stream past 192MB L2;
// uniform s_cbranch fast path for rows outside the freq band (~90% of rows).

typedef float v4f __attribute__((ext_vector_type(4)));

__global__ void __launch_bounds__(256) specaug_kernel(
    const v4f* __restrict__ x,
    const v4f* __restrict__ noise,
    const int* __restrict__ freq_start,
    const int* __restrict__ freq_width,
    const int* __restrict__ time_start,
    const int* __restrict__ time_width,
    const float* __restrict__ keep_mask,
    v4f* __restrict__ out)
{
    constexpr int   F     = 128;
    constexpr int   T     = 2048;
    constexpr float SCALE = 1.0f / (1.0f - 0.1f);   // 1/(1-DROPOUT_P)
    constexpr float NSTD  = 0.05f;

    const int blk = blockIdx.x;          // blk = b*F + f
    const int b   = blk >> 7;            // F == 128
    const int f   = blk & (F - 1);

    // Wave-uniform per-sample parameters -> scalar loads.
    const int  fs    = freq_start[b];
    const bool fband = (unsigned)(f - fs) < (unsigned)freq_width[b];
    const float drop = keep_mask[b] * SCALE;   // C == 1

    const long row = (long)blk * (T / 4);      // float4 index of row start

    if (!fband) {
        // Uniform fast path: factor == drop for the whole row.
#pragma unroll
        for (int k = 0; k < 2; ++k) {
            const int v = (int)threadIdx.x + k * 256;   // float4 slot [0,512)
            v4f xv = __builtin_nontemporal_load(&x[row + v]);
            v4f nv = __builtin_nontemporal_load(&noise[row + v]);
            v4f o;
#pragma unroll
            for (int e = 0; e < 4; ++e)
                o[e] = __builtin_fmaf(xv[e], drop, nv[e] * NSTD);
            __builtin_nontemporal_store(o, &out[row + v]);
        }
    } else {
        // Row intersects the freq band: apply the time-band zero mask.
        const int      t_lo = time_start[b];
        const unsigned t_w  = (unsigned)time_width[b];
#pragma unroll
        for (int k = 0; k < 2; ++k) {
            const int v = (int)threadIdx.x + k * 256;
            const int t = v * 4;
            v4f xv = __builtin_nontemporal_load(&x[row + v]);
            v4f nv = __builtin_nontemporal_load(&noise[row + v]);
            v4f o;
#pragma unroll
            for (int e = 0; e < 4; ++e) {
                const bool masked = (unsigned)(t + e - t_lo) < t_w;
                const float fac   = masked ? 0.0f : drop;
                o[e] = __builtin_fmaf(xv[e], fac, nv[e] * NSTD);
            }
            __builtin_nontemporal_store(o, &out[row + v]);
        }
    }
}

extern "C" void kernel_launch(void* const* d_in, const int* in_sizes, int n_in,
                              void* d_out, int out_size, void* d_ws, size_t ws_size,
                              hipStream_t stream) {
    // setup_inputs() order: x, freq_start, freq_width, time_start, time_width,
    //                       keep_mask, noise
    const v4f*   x     = (const v4f*)d_in[0];
    const int*   fs    = (const int*)d_in[1];
    const int*   fw    = (const int*)d_in[2];
    const int*   ts    = (const int*)d_in[3];
    const int*   tw    = (const int*)d_in[4];
    const float* keep  = (const float*)d_in[5];
    const v4f*   noise = (const v4f*)d_in[6];
    v4f*         out   = (v4f*)d_out;

    const int T    = 2048;
    const int rows = in_sizes[0] / T;   // B * F = 16384
    specaug_kernel<<<rows, 256, 0, stream>>>(x, noise, fs, fw, ts, tw, keep, out);
}